// BPRLoss_29463475651368
// MI455X (gfx1250) — compile-verified
//
#include <hip/hip_runtime.h>
#include <stdint.h>

// ---------------------------------------------------------------------------
// BPR loss:  nll = sum_b [ -mean_k( logsigmoid(pred[b,pos[b,k]] - pred[b,neg[b,k]]) ) ]
// predict (4096, 100000) f32; pos_idx/neg_idx (4096, 50) i64; output: 1 f32.
//
// Pure gather-latency workload (~50 MB of distinct lines, ~2 MFLOP): no matrix
// op exists here, so WMMA is inapplicable. The CDNA5-specific paths that do
// matter: async global->LDS staging (ASYNCcnt), wave32 shuffle reductions,
// split wait counters. 1 pair/thread -> 800 blocks = 6400 wave32 to keep the
// whole scatter stream in flight across every WGP.
// ---------------------------------------------------------------------------

namespace {
constexpr int kB     = 4096;
constexpr int kN     = 100000;
constexpr int kK     = 50;
constexpr int kPairs = kB * kK;          // 204800
constexpr int kBlock = 256;              // 8 wave32 per block
constexpr int kGrid1 = kPairs / kBlock;  // 800 blocks (exact)
}  // namespace

__device__ __forceinline__ float wave_reduce_add(float v) {
#pragma unroll
  for (int off = 16; off > 0; off >>= 1) {
    v += __shfl_down(v, off, 32);
  }
  return v;
}

__global__ __launch_bounds__(kBlock) void bpr_partial_kernel(
    const float* __restrict__ predict,
    const long long* __restrict__ pos_idx,
    const long long* __restrict__ neg_idx,
    float* __restrict__ partials) {
  __shared__ long long s_pos[kBlock];
  __shared__ long long s_neg[kBlock];
  __shared__ float s_wave[kBlock / 32];

  const int tid = threadIdx.x;
  const int i   = blockIdx.x * kBlock + tid;  // pair id, coalesced per block

  // ---- Stage the two 8-byte index loads via gfx1250 async global->LDS -----
  // Each lane copies into its own LDS slot and reads back only its own slot,
  // so a per-wave s_wait_asynccnt 0 is sufficient (no barrier needed).
  {
    unsigned lds_p = (unsigned)(uintptr_t)(const void*)&s_pos[tid];
    unsigned lds_n = (unsigned)(uintptr_t)(const void*)&s_neg[tid];
    unsigned long long ga_p = (unsigned long long)(uintptr_t)&pos_idx[i];
    unsigned long long ga_n = (unsigned long long)(uintptr_t)&neg_idx[i];
    asm volatile("global_load_async_to_lds_b64 %0, %1, off"
                 :: "v"(lds_p), "v"(ga_p) : "memory");
    asm volatile("global_load_async_to_lds_b64 %0, %1, off"
                 :: "v"(lds_n), "v"(ga_n) : "memory");
  }
  asm volatile("s_wait_asynccnt 0x0" ::: "memory");

  // ---- Two random gathers + stable logsigmoid -----------------------------
  const int b = i / kK;  // one small-constant division per thread
  const unsigned row = (unsigned)b * (unsigned)kN;
  const unsigned pi = (unsigned)s_pos[tid];
  const unsigned ni = (unsigned)s_neg[tid];
  const float pv = predict[(size_t)(row + pi)];
  const float nv = predict[(size_t)(row + ni)];
  const float x  = pv - nv;
  // logsigmoid(x) = min(x, 0) - log1p(exp(-|x|))   (numerically stable)
  const float acc = fminf(x, 0.0f) - log1pf(__expf(-fabsf(x)));

  // ---- Deterministic block reduction: wave32 shuffle tree + LDS -----------
  const int lane = tid & 31;
  const int wave = tid >> 5;
  float wsum = wave_reduce_add(acc);
  if (lane == 0) s_wave[wave] = wsum;
  __syncthreads();
  if (wave == 0) {
    float v = (lane < (kBlock / 32)) ? s_wave[lane] : 0.0f;
    v = wave_reduce_add(v);
    if (lane == 0) partials[blockIdx.x] = v;
  }
}

__global__ __launch_bounds__(kBlock) void bpr_final_kernel(
    const float* __restrict__ partials, float* __restrict__ out) {
  __shared__ float s_wave[kBlock / 32];
  const int tid  = threadIdx.x;
  const int lane = tid & 31;
  const int wave = tid >> 5;

  // Fixed-order strided accumulation over kGrid1 partials (deterministic).
  float v = 0.0f;
  for (int j = tid; j < kGrid1; j += kBlock) {
    v += partials[j];
  }
  v = wave_reduce_add(v);
  if (lane == 0) s_wave[wave] = v;
  __syncthreads();
  if (wave == 0) {
    float t = (lane < (kBlock / 32)) ? s_wave[lane] : 0.0f;
    t = wave_reduce_add(t);
    if (lane == 0) {
      // nll = -(1/K) * sum of logsigmoid terms
      out[0] = -t / (float)kK;
    }
  }
}

extern "C" void kernel_launch(void* const* d_in, const int* in_sizes, int n_in,
                              void* d_out, int out_size, void* d_ws, size_t ws_size,
                              hipStream_t stream) {
  (void)in_sizes; (void)n_in; (void)out_size; (void)ws_size;
  const float*     predict = (const float*)d_in[0];
  const long long* pos_idx = (const long long*)d_in[1];  // reference: int64
  const long long* neg_idx = (const long long*)d_in[2];  // reference: int64
  float* partials = (float*)d_ws;                        // kGrid1 floats

  bpr_partial_kernel<<<kGrid1, kBlock, 0, stream>>>(predict, pos_idx, neg_idx,
                                                    partials);
  bpr_final_kernel<<<1, kBlock, 0, stream>>>(partials, (float*)d_out);
}